// Attention_layer_38654705664405
// MI455X (gfx1250) — compile-verified
//
#include <hip/hip_runtime.h>
#include <hip/hip_bf16.h>

typedef __attribute__((ext_vector_type(2))) float v2f;
typedef __attribute__((ext_vector_type(8))) float v8f;

namespace {
constexpr int kB = 4, kC = 64, kN = 32, kH = 64, kW = 64;
constexpr int kS = 32, kM = 33, kCO = 61;        // M = N+1 tokens, CO = C-3
constexpr int kLdst = 50;                        // LDS row stride (floats), conflict-free
constexpr int kOffX = 0;                         // 64 x 50  (X tile, padded cols zeroed)
constexpr int kOffK = kOffX + 64 * kLdst;        // 32 x 50
constexpr int kOffQ = kOffK + 32 * kLdst;        // 32 x 50
constexpr int kOffV = kOffQ + 32 * kLdst;        // 32 x 50
constexpr int kOffA = kOffV + 32 * kLdst;        // 48 x 50  (scores/attn; reused for ao [61][32])
constexpr int kOffO = kOffA + 48 * kLdst;        // 32 x 50
constexpr int kWaveFloats = kOffO + 32 * kLdst;  // 12000 floats = 48 KB / wave
constexpr int kWavesPerWG = 4;
constexpr int kLdsBytes = kWavesPerWG * kWaveFloats * 4;  // 192 KB < 320 KB WGP LDS
}  // namespace

__device__ __forceinline__ v8f wmma4(v2f a, v2f b, v8f c) {
  // V_WMMA_F32_16X16X4_F32: D = A(16x4,f32) x B(4x16,f32) + C(16x16,f32)
  return __builtin_amdgcn_wmma_f32_16x16x4_f32(false, a, false, b, (short)0, c,
                                               false, false);
}

// ---------------------------------------------------------------------------
// Kernel 1: light positional embedding  le[b][s][i][j]  (B*32*33*33 floats)
// ---------------------------------------------------------------------------
__global__ void le_kernel(const float* __restrict__ light,
                          const float* __restrict__ lw1, const float* __restrict__ lb1,
                          const float* __restrict__ lw2, const float* __restrict__ lb2,
                          const float* __restrict__ lw3, const float* __restrict__ lb3,
                          float* __restrict__ le) {
  int t = blockIdx.x * blockDim.x + threadIdx.x;
  if (t >= kB * kM * kM) return;
  int b = t / (kM * kM);
  int r = t % (kM * kM);
  int i = r / kM, j = r % kM;

  float d[3];
#pragma unroll
  for (int c = 0; c < 3; ++c) {
    float li = (i < kN) ? light[(b * 3 + c) * kN + i] : -1.0f;
    float lj = (j < kN) ? light[(b * 3 + c) * kN + j] : -1.0f;
    d[c] = li - lj;
  }
  float h1[8];
#pragma unroll
  for (int o = 0; o < 8; ++o) {
    float v = lb1[o];
#pragma unroll
    for (int c = 0; c < 3; ++c) v += lw1[o * 3 + c] * d[c];
    h1[o] = (v > 0.0f) ? v : 0.1f * v;  // leaky_relu 0.1
  }
  float h2[16];
#pragma unroll
  for (int o = 0; o < 16; ++o) {
    float v = lb2[o];
#pragma unroll
    for (int c = 0; c < 8; ++c) v += lw2[o * 8 + c] * h1[c];
    h2[o] = (v > 0.0f) ? v : 0.1f * v;
  }
  for (int s = 0; s < kS; ++s) {
    float v = lb3[s];
#pragma unroll
    for (int c = 0; c < 16; ++c) v += lw3[s * 16 + c] * h2[c];
    le[((b * kS + s) * kM + i) * kM + j] = v;
  }
}

// One K/Q/V projection: dst[s,m] = Wm[s,:] @ sX[:,m] + bs[s].
// Inlined 3x with distinct constant-offset LDS pointers so all stores lower
// to ds_store (no generic-pointer phi -> no flat_store).
__device__ __forceinline__ void project(const float* __restrict__ Wm,
                                        const float* __restrict__ bs,
                                        const float* sX, float* dst,
                                        int half, int l16) {
#pragma unroll
  for (int st = 0; st < 2; ++st) {
    v8f acc0, acc1, acc2;
#pragma unroll
    for (int r = 0; r < 8; ++r) {
      float bb = bs[st * 16 + r + 8 * half];
      acc0[r] = bb; acc1[r] = bb; acc2[r] = bb;
    }
    for (int kk = 0; kk < 16; ++kk) {  // K-dim = C in steps of 4
      int c0 = kk * 4 + 2 * half;
      v2f a = {Wm[(st * 16 + l16) * kC + c0], Wm[(st * 16 + l16) * kC + c0 + 1]};
      v2f b0 = {sX[c0 * kLdst + l16], sX[(c0 + 1) * kLdst + l16]};
      v2f b1 = {sX[c0 * kLdst + 16 + l16], sX[(c0 + 1) * kLdst + 16 + l16]};
      v2f b2 = {sX[c0 * kLdst + 32 + l16], sX[(c0 + 1) * kLdst + 32 + l16]};
      acc0 = wmma4(a, b0, acc0);
      acc1 = wmma4(a, b1, acc1);
      acc2 = wmma4(a, b2, acc2);
    }
#pragma unroll
    for (int r = 0; r < 8; ++r) {
      int srow = (st * 16 + r + 8 * half) * kLdst;
      dst[srow + l16] = acc0[r];
      dst[srow + 16 + l16] = acc1[r];
      int m2 = 32 + l16;
      dst[srow + m2] = (m2 < kM) ? acc2[r] : 0.0f;  // zero pad tokens
    }
  }
}

// ---------------------------------------------------------------------------
// Kernel 2: fused attention. 1 wave = 1 pixel; 4 w-consecutive pixels per WG.
// ---------------------------------------------------------------------------
__global__ void __launch_bounds__(128) attn_kernel(
    const float* __restrict__ x, const float* __restrict__ le,
    const float* __restrict__ wk, const float* __restrict__ bk,
    const float* __restrict__ wq, const float* __restrict__ bq,
    const float* __restrict__ wv, const float* __restrict__ bv,
    const float* __restrict__ wo, const float* __restrict__ bo,
    float* __restrict__ out) {
  extern __shared__ float smem[];
  const int tid = threadIdx.x;
  const int wave = tid >> 5;
  const int lane = tid & 31;
  const int half = lane >> 4;  // WMMA lane-half
  const int l16 = lane & 15;

  const int P0 = blockIdx.x * kWavesPerWG;  // first pixel of this WG
  const int b = P0 >> 12;                   // 4096 pixels per batch (H*W)
  const int hh = (P0 >> 6) & 63;
  const int w0 = P0 & 63;                   // P0 % 4 == 0 -> w0..w0+3 in same row

  // ---- cooperative x staging via CDNA5 async global->LDS DMA --------------
  // GLOBAL_LOAD_ASYNC_TO_LDS_B32 (GV mode): copies a dword from global address gp to LDS byte offset ldsOff.
  // Dynamic LDS is the only LDS in this kernel -> smem starts at offset 0,
  // so computed byte offsets match the pointer-based layout below.
  {
    const int wsub = tid & 3;  // 4 threads per (c,n) pair -> 16B contiguity
    const int xbase = ((b * kC) * kN) * (kH * kW) + hh * kW + (w0 + wsub);
    const int ldsBaseB = (wsub * kWaveFloats + kOffX) * 4;
    for (int pair = tid >> 2; pair < kC * kN; pair += 32) {  // uniform trip count
      int c = pair >> 5, n = pair & 31;
      int ldsOff = ldsBaseB + (c * kLdst + n + 1) * 4;
      const float* gp = x + xbase + pair * (kH * kW);
      asm volatile("global_load_async_to_lds_b32 %0, %1, off"
                   :
                   : "v"(ldsOff), "v"(gp)
                   : "memory");
    }
    asm volatile("s_wait_asynccnt 0x0" ::: "memory");
  }
  __syncthreads();
  // ---- extra token (max over n) + zero the padded token columns ----------
  for (int task = tid; task < kWavesPerWG * kC; task += 128) {
    float* row = smem + (task >> 6) * kWaveFloats + kOffX + (task & 63) * kLdst;
    float mx = row[1];
    for (int n = 1; n < kN; ++n) mx = fmaxf(mx, row[1 + n]);
    row[0] = mx;
    for (int m = kM; m < 48; ++m) row[m] = 0.0f;
  }
  __syncthreads();

  float* sX = smem + wave * kWaveFloats + kOffX;
  float* sK = smem + wave * kWaveFloats + kOffK;
  float* sQ = smem + wave * kWaveFloats + kOffQ;
  float* sV = smem + wave * kWaveFloats + kOffV;
  float* sA = smem + wave * kWaveFloats + kOffA;
  float* sO = smem + wave * kWaveFloats + kOffO;

  // ---- K/Q/V projections (fp32 WMMA), three explicit instantiations ------
  project(wk, bk, sX, sK, half, l16);
  project(wq, bq, sX, sQ, half, l16);
  project(wv, bv, sX, sV, half, l16);

  // ---- scores[i,j] = scale * sum_s kt[s,i]*qt[s,j] ------------------------
  const float scale = 0.17677669529663687f;  // 1/sqrt(32)
  for (int it = 0; it < 3; ++it)
    for (int jt = 0; jt < 3; ++jt) {
      v8f acc = {0, 0, 0, 0, 0, 0, 0, 0};
      for (int kk = 0; kk < 8; ++kk) {
        int s0 = kk * 4 + 2 * half;
        v2f a = {sK[s0 * kLdst + it * 16 + l16], sK[(s0 + 1) * kLdst + it * 16 + l16]};
        v2f bq2 = {sQ[s0 * kLdst + jt * 16 + l16], sQ[(s0 + 1) * kLdst + jt * 16 + l16]};
        acc = wmma4(a, bq2, acc);
      }
#pragma unroll
      for (int r = 0; r < 8; ++r)
        sA[(it * 16 + r + 8 * half) * kLdst + jt * 16 + l16] = acc[r] * scale;
    }

  // ---- softmax over key index i, per column j -----------------------------
#pragma unroll
  for (int base = 0; base < 64; base += 32) {
    int j = base + lane;
    if (j < kM) {
      float mx = -1e30f;
      for (int i = 0; i < kM; ++i) mx = fmaxf(mx, sA[i * kLdst + j]);
      float sum = 0.0f;
      for (int i = 0; i < kM; ++i) {
        float e = __expf(sA[i * kLdst + j] - mx);
        sA[i * kLdst + j] = e;
        sum += e;
      }
      float inv = 1.0f / sum;
      for (int i = 0; i < kM; ++i) sA[i * kLdst + j] *= inv;
    }
  }

  // ---- additive light bias: a[k,g] += sum_s qt[s,k]*le[b,s,k,g] -----------
  {
    const float* leb = le + b * (kS * kM * kM);
#pragma unroll
    for (int base = 0; base < 64; base += 32) {
      int g = base + lane;  // coalesced in g across lanes; pure L2 hits
      if (g < kM) {
        for (int k = 0; k < kM; ++k) {
          const float* lp = leb + k * kM + g;
          float acc = 0.0f;
          for (int s = 0; s < kS; ++s)
            acc += sQ[s * kLdst + k] * lp[s * (kM * kM)];
          sA[k * kLdst + g] += acc;
        }
      }
    }
  }

  // ---- o[s,g] = sum_m vt[s,m]*a[m,g]  (vt pad cols are exact zeros) -------
  for (int st = 0; st < 2; ++st)
    for (int gt = 0; gt < 3; ++gt) {
      v8f acc = {0, 0, 0, 0, 0, 0, 0, 0};
      for (int kk = 0; kk < 12; ++kk) {  // K-dim = m in 0..47
        int m0 = kk * 4 + 2 * half;
        v2f a = {sV[(st * 16 + l16) * kLdst + m0], sV[(st * 16 + l16) * kLdst + m0 + 1]};
        v2f ba = {sA[m0 * kLdst + gt * 16 + l16], sA[(m0 + 1) * kLdst + gt * 16 + l16]};
        acc = wmma4(a, ba, acc);
      }
#pragma unroll
      for (int r = 0; r < 8; ++r)
        sO[(st * 16 + r + 8 * half) * kLdst + gt * 16 + l16] = acc[r];
    }

  // ---- ao[oc,g] = wo @ o + bo; keep g=1..32 into sAo (reuses sA region) ---
  float* sAo = sA;  // [61][32], stride 32 (1952 <= 2400 floats)
  for (int ot = 0; ot < 4; ++ot)
    for (int gt = 0; gt < 3; ++gt) {
      v8f acc;
#pragma unroll
      for (int r = 0; r < 8; ++r) {
        int oc = ot * 16 + r + 8 * half;
        acc[r] = (oc < kCO) ? bo[oc < kCO ? oc : kCO - 1] : 0.0f;
      }
      int ocA = ot * 16 + l16;
      int ocAc = ocA < kCO ? ocA : kCO - 1;     // clamp: in-bounds load
      float amask = (ocA < kCO) ? 1.0f : 0.0f;  // branchless mask (no EXEC change)
      for (int kk = 0; kk < 8; ++kk) {
        int s0 = kk * 4 + 2 * half;
        v2f a = {wo[ocAc * kS + s0] * amask, wo[ocAc * kS + s0 + 1] * amask};
        v2f bO = {sO[s0 * kLdst + gt * 16 + l16], sO[(s0 + 1) * kLdst + gt * 16 + l16]};
        acc = wmma4(a, bO, acc);
      }
      int g = gt * 16 + l16;
#pragma unroll
      for (int r = 0; r < 8; ++r) {
        int oc = ot * 16 + r + 8 * half;
        if (oc < kCO && g >= 1 && g < kM) sAo[oc * 32 + (g - 1)] = acc[r];
      }
    }

  __syncthreads();
  // ---- cooperative residual store (drops token 0) -------------------------
  {
    const int wsub = tid & 3;
    const int obase = ((b * kC) * kN) * (kH * kW) + hh * kW + (w0 + wsub);
    const float* sXp = smem + wsub * kWaveFloats + kOffX;
    const float* sAp = smem + wsub * kWaveFloats + kOffA;
    for (int pair = tid >> 2; pair < kC * kN; pair += 32) {
      int ch = pair >> 5, n = pair & 31;
      float v = sXp[ch * kLdst + n + 1];  // original x (token n = m-1)
      int oc = ch - 3;
      float add = (oc >= 0) ? sAp[oc * 32 + n] : 0.0f;
      out[obase + pair * (kH * kW)] = v + add;
    }
  }
}

// ---------------------------------------------------------------------------
extern "C" void kernel_launch(void* const* d_in, const int* in_sizes, int n_in,
                              void* d_out, int out_size, void* d_ws, size_t ws_size,
                              hipStream_t stream) {
  (void)in_sizes; (void)n_in; (void)out_size; (void)ws_size;
  const float* x = (const float*)d_in[0];
  const float* light = (const float*)d_in[1];
  const float* wk = (const float*)d_in[2];
  const float* bk = (const float*)d_in[3];
  const float* wq = (const float*)d_in[4];
  const float* bq = (const float*)d_in[5];
  const float* wv = (const float*)d_in[6];
  const float* bv = (const float*)d_in[7];
  const float* wo = (const float*)d_in[8];
  const float* bo = (const float*)d_in[9];
  const float* lw1 = (const float*)d_in[10];
  const float* lb1 = (const float*)d_in[11];
  const float* lw2 = (const float*)d_in[12];
  const float* lb2 = (const float*)d_in[13];
  const float* lw3 = (const float*)d_in[14];
  const float* lb3 = (const float*)d_in[15];
  float* out = (float*)d_out;
  float* le = (float*)d_ws;  // B*S*M*M floats = 557,568 bytes of scratch

  int leTasks = kB * kM * kM;
  le_kernel<<<(leTasks + 127) / 128, 128, 0, stream>>>(light, lw1, lb1, lw2, lb2,
                                                       lw3, lb3, le);

  (void)hipFuncSetAttribute((const void*)attn_kernel,
                            hipFuncAttributeMaxDynamicSharedMemorySize, kLdsBytes);
  int blocks = (kB * kH * kW) / kWavesPerWG;  // 4096 WGs x 128 threads
  attn_kernel<<<blocks, 128, kLdsBytes, stream>>>(x, le, wk, bk, wq, bq, wv, bv,
                                                  wo, bo, out);
}